// MultiHeadAttentionConv_34471407517996
// MI455X (gfx1250) — compile-verified
//
#include <hip/hip_runtime.h>
#include <hip/hip_bf16.h>
#include <math.h>

#define NN 50000
#define EE 800000
#define DIM 128
#define HH 8
#define CC 16
#define EDIM 32
#define OUTD 128   // H*C
#define SROW 132   // padded LDS row stride (floats): banks (4*mr + k) % 64 distinct per lane

typedef __attribute__((ext_vector_type(2))) float v2f;
typedef __attribute__((ext_vector_type(8))) float v8f;
typedef __attribute__((vector_size(4 * sizeof(int)))) int v4i;

#if __has_builtin(__builtin_amdgcn_global_load_async_to_lds_b128)
#define HAVE_ASYNC_LDS 1
typedef __attribute__((address_space(3))) v4i lds_v4i;
#else
#define HAVE_ASYNC_LDS 0
#endif

// ---------------------------------------------------------------------------
// Kernel 0: init. Zero d_out, zero denom, set segment-max to -inf.
// ---------------------------------------------------------------------------
__global__ void k_init(float* out, float* m, float* denom) {
    int idx = blockIdx.x * blockDim.x + threadIdx.x;
    if (idx < NN * OUTD) out[idx] = 0.0f;
    if (idx < NN * HH) {
        m[idx] = -INFINITY;
        denom[idx] = 0.0f;
    }
}

// ---------------------------------------------------------------------------
// Kernel 1: QKV projection via f32 WMMA 16x16x4.
//   grid.x = 3125 row-tiles (N = 3125*16 exactly), grid.y = 3 (q,k,v)
//   block  = 256 threads = 8 wave32s; wave w computes column tile n0 = w*16.
//   The 16x128 A-stripe (contiguous, 8 KB) is staged once per block into LDS
//   via async-to-LDS (ASYNCcnt) when available; all 8 waves then read A
//   fragments from LDS (padded stride kills bank conflicts).
// ---------------------------------------------------------------------------
__global__ __launch_bounds__(256) void k_qkv_wmma(const float* __restrict__ x,
                                                  const float* __restrict__ Wq,
                                                  const float* __restrict__ Wk,
                                                  const float* __restrict__ Wv,
                                                  float* __restrict__ qkv) {
    __shared__ float sA[16 * SROW];

    const int lane = threadIdx.x & 31;
    const int wave = threadIdx.x >> 5;

    const int m0 = blockIdx.x * 16;
    const int n0 = wave * 16;
    const int which = blockIdx.y;  // 0=q, 1=k, 2=v

    const float* __restrict__ W = (which == 0) ? Wq : (which == 1) ? Wk : Wv;
    float* __restrict__ dst = qkv + (size_t)which * NN * OUTD;

    // --- stage x[m0..m0+15][0..127] into LDS (512 float4 chunks, 2 per thread)
    const float* __restrict__ xs = x + (size_t)m0 * DIM;  // contiguous 2048 floats
    #pragma unroll
    for (int it = 0; it < 2; ++it) {
        const int c = threadIdx.x + it * 256;   // chunk id 0..511
        const int row = c >> 5;                 // 0..15
        const int col4 = (c & 31) << 2;         // 0..124 step 4
#if HAVE_ASYNC_LDS
        __builtin_amdgcn_global_load_async_to_lds_b128(
            (v4i*)(xs + row * DIM + col4),
            (lds_v4i*)(&sA[row * SROW + col4]), 0, 0);
#else
        *(float4*)(&sA[row * SROW + col4]) = *(const float4*)(xs + row * DIM + col4);
#endif
    }
#if HAVE_ASYNC_LDS
#if __has_builtin(__builtin_amdgcn_s_wait_asynccnt)
    __builtin_amdgcn_s_wait_asynccnt(0);
#else
    asm volatile("s_wait_asynccnt 0" ::: "memory");
#endif
#endif
    __syncthreads();

    const int mr = lane & 15;           // row within A tile
    const int kh = (lane >> 4) << 1;    // K sub-offset: 0 or 2
    const int nc = lane & 15;           // col within B/D tile

    v8f acc = {};
    #pragma unroll
    for (int k0 = 0; k0 < DIM; k0 += 4) {
        v2f a, b;
        a.x = sA[mr * SROW + (k0 + kh)];
        a.y = sA[mr * SROW + (k0 + kh + 1)];
        b.x = W[(size_t)(k0 + kh) * OUTD + (n0 + nc)];
        b.y = W[(size_t)(k0 + kh + 1) * OUTD + (n0 + nc)];
        acc = __builtin_amdgcn_wmma_f32_16x16x4_f32(
            /*neg_a=*/false, a, /*neg_b=*/false, b,
            /*c_mod=*/(short)0, acc, /*reuse_a=*/false, /*reuse_b=*/false);
    }

    const int row_hi = (lane >> 4) << 3;  // 0 or 8
    #pragma unroll
    for (int r = 0; r < 8; ++r) {
        dst[(size_t)(m0 + r + row_hi) * OUTD + (n0 + nc)] = acc[r];
    }
}

// ---------------------------------------------------------------------------
// atomic float max via int/uint punning (works for mixed signs, init = -inf)
// ---------------------------------------------------------------------------
__device__ __forceinline__ void atomicMaxFloat(float* addr, float val) {
    if (val >= 0.0f) {
        atomicMax((int*)addr, __float_as_int(val));
    } else {
        atomicMin((unsigned int*)addr, (unsigned int)__float_as_int(val));
    }
}

// ---------------------------------------------------------------------------
// Kernel 2: per-(edge,head) attention logits + segment max.
//   score[e,h] = dot(q[dst,h,:], k[src,h,:]) / sqrt(C) + edge_attr[e,:] @ We[:,h]
// ---------------------------------------------------------------------------
__global__ __launch_bounds__(256) void k_scores(const int* __restrict__ edge_index,
                                                const float* __restrict__ edge_attr,
                                                const float* __restrict__ We,
                                                const float* __restrict__ qkv,
                                                float* __restrict__ scores,
                                                float* __restrict__ m) {
    int idx = blockIdx.x * blockDim.x + threadIdx.x;
    if (idx >= EE * HH) return;
    const int e = idx >> 3;       // H == 8
    const int h = idx & 7;

    const int src = edge_index[e];
    const int dst = edge_index[EE + e];

    const float* __restrict__ q = qkv + (size_t)dst * OUTD + h * CC;
    const float* __restrict__ k = qkv + (size_t)NN * OUTD + (size_t)src * OUTD + h * CC;

    __builtin_prefetch(edge_attr + (size_t)(e + 8) * EDIM, 0, 1);

    float dot = 0.0f;
    #pragma unroll
    for (int c = 0; c < CC; ++c) dot += q[c] * k[c];

    float ed = 0.0f;
    const float* __restrict__ ea = edge_attr + (size_t)e * EDIM;
    #pragma unroll
    for (int d = 0; d < EDIM; ++d) ed += ea[d] * We[d * HH + h];

    const float s = dot * 0.25f + ed;  // 1/sqrt(16) = 0.25
    scores[idx] = s;
    atomicMaxFloat(&m[dst * HH + h], s);
}

// ---------------------------------------------------------------------------
// Kernel 3 (fused): e = exp(score - m[dst]); accumulate denom AND e * v[src]
// into out. Normalization happens per-node in kernel 4 — saves a full edge
// pass versus computing alpha explicitly.
// ---------------------------------------------------------------------------
__global__ __launch_bounds__(256) void k_accum(const int* __restrict__ edge_index,
                                               const float* __restrict__ m,
                                               const float* __restrict__ scores,
                                               const float* __restrict__ qkv,
                                               float* __restrict__ denom,
                                               float* __restrict__ out) {
    int idx = blockIdx.x * blockDim.x + threadIdx.x;
    if (idx >= EE * HH) return;
    const int e = idx >> 3;
    const int h = idx & 7;
    const int src = edge_index[e];
    const int dst = edge_index[EE + e];

    const float ex = __expf(scores[idx] - m[dst * HH + h]);
    atomicAdd(&denom[dst * HH + h], ex);

    const float* __restrict__ v = qkv + (size_t)2 * NN * OUTD + (size_t)src * OUTD + h * CC;
    float* __restrict__ o = out + (size_t)dst * OUTD + h * CC;
    #pragma unroll
    for (int c = 0; c < CC; ++c) {
        atomicAdd(&o[c], v[c] * ex);
    }
}

// ---------------------------------------------------------------------------
// Kernel 4: per-node normalization: out[n,h,c] /= (denom[n,h] + 1e-16)
// ---------------------------------------------------------------------------
__global__ __launch_bounds__(256) void k_norm(const float* __restrict__ denom,
                                              float* __restrict__ out) {
    int idx = blockIdx.x * blockDim.x + threadIdx.x;
    if (idx >= NN * OUTD) return;
    const int n = idx >> 7;             // / 128
    const int h = (idx >> 4) & 7;       // head
    out[idx] = out[idx] / (denom[n * HH + h] + 1e-16f);
}

// ---------------------------------------------------------------------------
// Launch
// ---------------------------------------------------------------------------
extern "C" void kernel_launch(void* const* d_in, const int* in_sizes, int n_in,
                              void* d_out, int out_size, void* d_ws, size_t ws_size,
                              hipStream_t stream) {
    const float* x          = (const float*)d_in[0];
    const int*   edge_index = (const int*)d_in[1];
    const float* edge_attr  = (const float*)d_in[2];
    const float* Wq         = (const float*)d_in[3];
    const float* Wk         = (const float*)d_in[4];
    const float* Wv         = (const float*)d_in[5];
    const float* We         = (const float*)d_in[6];
    float* out = (float*)d_out;

    // workspace layout (floats)
    float* ws     = (float*)d_ws;
    float* qkv    = ws;                                   // 3 * N * 128
    float* scores = qkv + (size_t)3 * NN * OUTD;          // E * H
    float* m      = scores + (size_t)EE * HH;             // N * H
    float* denom  = m + (size_t)NN * HH;                  // N * H

    // init
    {
        int total = NN * OUTD;
        k_init<<<(total + 255) / 256, 256, 0, stream>>>(out, m, denom);
    }

    // QKV projection: 3125 row tiles x 3 matrices, 8 waves/block = 8 col tiles
    {
        dim3 grid(NN / 16, 3, 1);
        k_qkv_wmma<<<grid, 256, 0, stream>>>(x, Wq, Wk, Wv, qkv);
    }

    // scores + segment max
    {
        int total = EE * HH;
        k_scores<<<(total + 255) / 256, 256, 0, stream>>>(edge_index, edge_attr, We,
                                                          qkv, scores, m);
    }

    // fused exp + denom + weighted scatter
    {
        int total = EE * HH;
        k_accum<<<(total + 255) / 256, 256, 0, stream>>>(edge_index, m, scores,
                                                         qkv, denom, out);
    }

    // per-node normalization
    {
        int total = NN * OUTD;
        k_norm<<<(total + 255) / 256, 256, 0, stream>>>(denom, out);
    }
}